// RGCN_29420525978184
// MI455X (gfx1250) — compile-verified
//
#include <hip/hip_runtime.h>
#include <hip/hip_bf16.h>

typedef __attribute__((ext_vector_type(2))) float v2f;
typedef __attribute__((ext_vector_type(8))) float v8f;

#define NNODES 50000
#define NREL   3
#define NEDGE  600000
#define FIN    128
#define FHID   64
#define FOUT   64
#define MTILES (NNODES / 16)   // 3125, exact

// ---------------------------------------------------------------- utilities
__global__ void zero_f4(float4* __restrict__ p, int n4) {
    int i = blockIdx.x * blockDim.x + threadIdx.x;
    if (i < n4) p[i] = make_float4(0.f, 0.f, 0.f, 0.f);
}

__global__ void degrees_kernel(const int* __restrict__ src, const int* __restrict__ dst,
                               float* __restrict__ degO, float* __restrict__ degI) {
    int t = blockIdx.x * blockDim.x + threadIdx.x;
    int r = blockIdx.y;
    if (t >= NEDGE) return;
    int s = src[(size_t)r * NEDGE + t];
    int d = dst[(size_t)r * NEDGE + t];
    unsafeAtomicAdd(&degO[r * NNODES + s], 1.0f);
    unsafeAtomicAdd(&degI[r * NNODES + d], 1.0f);
}

__global__ void invsqrt_deg(float* __restrict__ deg, int n) {
    int i = blockIdx.x * blockDim.x + threadIdx.x;
    if (i < n) deg[i] = rsqrtf(fmaxf(deg[i], 1.0f));
}

// ------------------------------------------------- layer-1 GEMM (fp32 WMMA)
// h1[r] = (x * invOut[r][row]) @ W1[r]   : [N,128] x [128,64]
__global__ __launch_bounds__(128) void gemm_layer1(
    const float* __restrict__ X, const float* __restrict__ W1,
    const float* __restrict__ invOut, float* __restrict__ H1) {
    __shared__ float Wlds[FIN * FHID];      // 32 KB
    __shared__ float Alds[4][16 * FIN];     // 32 KB (per-wave A tiles)
    const int tid  = threadIdx.x;
    const int wave = tid >> 5, lane = tid & 31;
    const int r = blockIdx.y;
    // cooperative load of W1[r] into LDS
    const float4* Wg = (const float4*)(W1 + (size_t)r * FIN * FHID);
    for (int i = tid; i < FIN * FHID / 4; i += 128) ((float4*)Wlds)[i] = Wg[i];
    const int tileM = blockIdx.x * 4 + wave;
    if (tileM < MTILES) {
        const int m0 = tileM * 16;
        // stage 16x128 A tile, scaled by src-side norm (row = i, f4-col = lane)
        for (int i = 0; i < 16; ++i) {
            float s = invOut[r * NNODES + m0 + i];
            float4 v = ((const float4*)(X + (size_t)(m0 + i) * FIN))[lane];
            v.x *= s; v.y *= s; v.z *= s; v.w *= s;
            ((float4*)&Alds[wave][i * FIN])[lane] = v;
        }
    }
    __syncthreads();
    if (tileM < MTILES) {
        const int row16 = lane & 15, half = lane >> 4;
        const float* Aw = Alds[wave];
        v8f acc[4] = {};
#pragma unroll 8
        for (int k = 0; k < FIN; k += 4) {
            v2f a;
            a.x = Aw[row16 * FIN + k + half * 2];
            a.y = Aw[row16 * FIN + k + half * 2 + 1];
#pragma unroll
            for (int nt = 0; nt < 4; ++nt) {
                v2f b;
                b.x = Wlds[(k + half * 2) * FHID + nt * 16 + row16];
                b.y = Wlds[(k + half * 2 + 1) * FHID + nt * 16 + row16];
                acc[nt] = __builtin_amdgcn_wmma_f32_16x16x4_f32(
                    false, a, false, b, (short)0, acc[nt], false, false);
            }
        }
        float* out = H1 + ((size_t)r * NNODES + tileM * 16) * FHID;
#pragma unroll
        for (int nt = 0; nt < 4; ++nt)
#pragma unroll
            for (int i = 0; i < 8; ++i)
                out[(i + half * 8) * FHID + nt * 16 + row16] = acc[nt][i];
    }
}

// --------------------------------------------------------------- scatters
__global__ void scatter1(const int* __restrict__ src, const int* __restrict__ dst,
                         const float* __restrict__ H1, float* __restrict__ AGG1) {
    int t = blockIdx.x * blockDim.x + threadIdx.x;    // NEDGE*16 per relation
    int r = blockIdx.y;
    int e = t >> 4, c4 = (t & 15) * 4;
    if (e >= NEDGE) return;
    int s = src[(size_t)r * NEDGE + e];
    int d = dst[(size_t)r * NEDGE + e];
    float4 v = *(const float4*)(H1 + ((size_t)r * NNODES + s) * FHID + c4);
    float* ap = AGG1 + ((size_t)r * NNODES + d) * FHID + c4;
    unsafeAtomicAdd(ap + 0, v.x);
    unsafeAtomicAdd(ap + 1, v.y);
    unsafeAtomicAdd(ap + 2, v.z);
    unsafeAtomicAdd(ap + 3, v.w);
}

__global__ void scatter2(const int* __restrict__ src, const int* __restrict__ dst,
                         const float* __restrict__ H, const float* __restrict__ invOut,
                         float* __restrict__ AGG2) {
    int t = blockIdx.x * blockDim.x + threadIdx.x;
    int r = blockIdx.y;
    int e = t >> 4, c4 = (t & 15) * 4;
    if (e >= NEDGE) return;
    int s = src[(size_t)r * NEDGE + e];
    int d = dst[(size_t)r * NEDGE + e];
    float sc = invOut[r * NNODES + s];
    float4 v = *(const float4*)(H + (size_t)s * FHID + c4);
    float* ap = AGG2 + ((size_t)r * NNODES + d) * FHID + c4;
    unsafeAtomicAdd(ap + 0, v.x * sc);
    unsafeAtomicAdd(ap + 1, v.y * sc);
    unsafeAtomicAdd(ap + 2, v.z * sc);
    unsafeAtomicAdd(ap + 3, v.w * sc);
}

// ------------------------------------------- combine relations + bias + ReLU
__global__ void combine1(const float* __restrict__ AGG1, const float* __restrict__ invIn,
                         const float* __restrict__ b1, float* __restrict__ H) {
    int idx = blockIdx.x * blockDim.x + threadIdx.x;
    if (idx >= NNODES * FHID) return;
    int n = idx >> 6, c = idx & 63;
    float v = 0.f;
#pragma unroll
    for (int r = 0; r < NREL; ++r)
        v += AGG1[((size_t)r * NNODES + n) * FHID + c] * invIn[r * NNODES + n]
           + b1[r * FHID + c];
    H[idx] = fmaxf(v, 0.f);
}

// --------------------------------- layer-2 GEMM, summed over relations
// out = sum_r (AGG2[r] * invIn[r][row]) @ W2[r] + sum_r b2[r]
__global__ __launch_bounds__(128) void gemm_layer2(
    const float* __restrict__ AGG2, const float* __restrict__ W2,
    const float* __restrict__ invIn, const float* __restrict__ b2,
    float* __restrict__ OUT) {
    __shared__ float Wlds[FHID * FOUT];     // 16 KB
    __shared__ float Alds[4][16 * FHID];    // 16 KB
    const int tid  = threadIdx.x;
    const int wave = tid >> 5, lane = tid & 31;
    const int tileM = blockIdx.x * 4 + wave;
    const int row16 = lane & 15, half = lane >> 4;
    v8f acc[4] = {};
    for (int r = 0; r < NREL; ++r) {
        __syncthreads();   // previous round done reading LDS
        const float4* Wg = (const float4*)(W2 + (size_t)r * FHID * FOUT);
        for (int i = tid; i < FHID * FOUT / 4; i += 128) ((float4*)Wlds)[i] = Wg[i];
        if (tileM < MTILES) {
            const int m0 = tileM * 16;
            // 16x64 tile: row = 2i + lane/16, f4-col = lane%16
            for (int i = 0; i < 8; ++i) {
                int row = i * 2 + half, c4 = row16;
                float s = invIn[r * NNODES + m0 + row];
                float4 v = ((const float4*)(AGG2 + ((size_t)r * NNODES + m0 + row) * FHID))[c4];
                v.x *= s; v.y *= s; v.z *= s; v.w *= s;
                ((float4*)&Alds[wave][row * FHID])[c4] = v;
            }
        }
        __syncthreads();
        if (tileM < MTILES) {
            const float* Aw = Alds[wave];
#pragma unroll 8
            for (int k = 0; k < FHID; k += 4) {
                v2f a;
                a.x = Aw[row16 * FHID + k + half * 2];
                a.y = Aw[row16 * FHID + k + half * 2 + 1];
#pragma unroll
                for (int nt = 0; nt < 4; ++nt) {
                    v2f b;
                    b.x = Wlds[(k + half * 2) * FOUT + nt * 16 + row16];
                    b.y = Wlds[(k + half * 2 + 1) * FOUT + nt * 16 + row16];
                    acc[nt] = __builtin_amdgcn_wmma_f32_16x16x4_f32(
                        false, a, false, b, (short)0, acc[nt], false, false);
                }
            }
        }
    }
    if (tileM < MTILES) {
        const int m0 = tileM * 16;
#pragma unroll
        for (int nt = 0; nt < 4; ++nt) {
            int col = nt * 16 + row16;
            float bs = b2[col] + b2[FOUT + col] + b2[2 * FOUT + col];
#pragma unroll
            for (int i = 0; i < 8; ++i)
                OUT[(size_t)(m0 + i + half * 8) * FOUT + col] = acc[nt][i] + bs;
        }
    }
}

// ------------------------------------------------------------------ driver
extern "C" void kernel_launch(void* const* d_in, const int* in_sizes, int n_in,
                              void* d_out, int out_size, void* d_ws, size_t ws_size,
                              hipStream_t stream) {
    (void)in_sizes; (void)n_in; (void)out_size; (void)ws_size;
    const float* x  = (const float*)d_in[0];
    const float* W1 = (const float*)d_in[1];
    const float* b1 = (const float*)d_in[2];
    const float* W2 = (const float*)d_in[3];
    const float* b2 = (const float*)d_in[4];
    const int*  src = (const int*)d_in[5];
    const int*  dst = (const int*)d_in[6];
    float* out = (float*)d_out;

    float* ws = (float*)d_ws;
    const size_t RN = (size_t)NREL * NNODES;
    float* degO = ws;               // R*N  -> becomes invOut (rsqrt in place)
    float* degI = ws + RN;          // R*N  -> becomes invIn
    float* B1   = ws + 2 * RN;      // R*N*64 : h1, later reused as agg2
    float* B2   = B1 + RN * FHID;   // R*N*64 : agg1
    float* H    = B2 + RN * FHID;   // N*64   : post-ReLU hidden

    // 1. zero degrees (both arrays contiguous) + agg1
    zero_f4<<<(int)((2 * RN / 4 + 255) / 256), 256, 0, stream>>>((float4*)degO, (int)(2 * RN / 4));
    zero_f4<<<(int)((RN * FHID / 4 + 255) / 256), 256, 0, stream>>>((float4*)B2, (int)(RN * FHID / 4));
    // 2. degree histograms (shared by both layers)
    degrees_kernel<<<dim3((NEDGE + 255) / 256, NREL), 256, 0, stream>>>(src, dst, degO, degI);
    // 3. deg -> rsqrt(max(deg,1)) in place
    invsqrt_deg<<<(int)((2 * RN + 255) / 256), 256, 0, stream>>>(degO, (int)(2 * RN));
    // 4. layer-1 dense GEMM (src-norm fused)
    gemm_layer1<<<dim3((MTILES + 3) / 4, NREL), 128, 0, stream>>>(x, W1, degO, B1);
    // 5. edge scatter, layer 1
    scatter1<<<dim3(NEDGE * 16 / 256, NREL), 256, 0, stream>>>(src, dst, B1, B2);
    // 6. in-norm + bias + relation-sum + ReLU
    combine1<<<(NNODES * FHID + 255) / 256, 256, 0, stream>>>(B2, degI, b1, H);
    // 7. zero agg2 (reuse h1 buffer)
    zero_f4<<<(int)((RN * FHID / 4 + 255) / 256), 256, 0, stream>>>((float4*)B1, (int)(RN * FHID / 4));
    // 8. edge scatter, layer 2 (src-norm fused)
    scatter2<<<dim3(NEDGE * 16 / 256, NREL), 256, 0, stream>>>(src, dst, H, degO, B1);
    // 9. layer-2 GEMM over all relations, in-norm + Σbias fused, writes d_out
    gemm_layer2<<<(MTILES + 3) / 4, 128, 0, stream>>>(B1, W2, degI, b2, out);
}